// Center_Loss_22900765622964
// MI455X (gfx1250) — compile-verified
//
#include <hip/hip_runtime.h>

typedef __attribute__((ext_vector_type(2))) float v2f;
typedef __attribute__((ext_vector_type(8))) float v8f;

#define B_SZ   4096
#define C_SZ   16384
#define F_SZ   512
#define TILE   16                 // rows of x per block
#define THREADS 256
#define WAVES   8                 // 256 / wave32
#define K_PER_WAVE (F_SZ / WAVES) // 64
#define CHUNKS_PER_WAVE (K_PER_WAVE / 4) // 16 WMMA k-steps per wave

// Kernel 1: one block = 16 rows.  Gather centers[labels[b]] rows into LDS,
// run fp32 WMMA 16x16x4 over K to get Gram-diagonals (x.x, c.c, x.c), then
// per-row clamp and a per-block partial sum.
__global__ __launch_bounds__(THREADS)
void center_loss_tile_kernel(const float* __restrict__ x,
                             const int* __restrict__ labels,
                             const float* __restrict__ centers,
                             float* __restrict__ partials) {
    __shared__ float smem[2 * TILE * F_SZ];   // 64 KB: Xs | Cs
    float* Xs = smem;
    float* Cs = smem + TILE * F_SZ;

    const int tid = threadIdx.x;
    const int b0  = blockIdx.x * TILE;

    // ---- coalesced global -> LDS: x tile (contiguous rows) ----
    {
        const float4* src = (const float4*)(x + (size_t)b0 * F_SZ);
        float4*       dst = (float4*)Xs;
        #pragma unroll
        for (int i = tid; i < TILE * F_SZ / 4; i += THREADS)
            dst[i] = src[i];
    }
    // ---- gathered center rows (labels are L2-hot; centers fit in 192MB L2) ----
    {
        float4* dst = (float4*)Cs;
        #pragma unroll
        for (int i = tid; i < TILE * F_SZ / 4; i += THREADS) {
            const int r   = i >> 7;      // / (F_SZ/4 = 128 float4 per row)
            const int c   = i & 127;
            const int lab = labels[b0 + r];
            dst[i] = ((const float4*)(centers + (size_t)lab * F_SZ))[c];
        }
    }
    __syncthreads();

    const int wave = tid >> 5;
    const int lane = tid & 31;
    const int row  = lane & 15;               // A-frag: lane L holds row L%16
    const int koff = (lane >> 4) << 1;        // lanes 0-15: K+0,+1 ; 16-31: K+2,+3

    v8f acc_xx = {};   // diag -> sum x^2
    v8f acc_cc = {};   // diag -> sum c^2
    v8f acc_xc = {};   // diag -> x . c
    const int kbase = wave * K_PER_WAVE;
    #pragma unroll
    for (int ck = 0; ck < CHUNKS_PER_WAVE; ++ck) {
        const int k = kbase + ck * 4;
        // 8B-aligned LDS loads; A-frag of M and B-frag of M^T share register layout
        v2f a = *(const v2f*)(&Xs[row * F_SZ + k + koff]);
        v2f b = *(const v2f*)(&Cs[row * F_SZ + k + koff]);
        acc_xx = __builtin_amdgcn_wmma_f32_16x16x4_f32(false, a, false, a,
                                                       (short)0, acc_xx, false, false);
        acc_cc = __builtin_amdgcn_wmma_f32_16x16x4_f32(false, b, false, b,
                                                       (short)0, acc_cc, false, false);
        acc_xc = __builtin_amdgcn_wmma_f32_16x16x4_f32(false, a, false, b,
                                                       (short)0, acc_xc, false, false);
    }

    __syncthreads();   // all WMMA LDS reads done -> reuse smem for reduction

    float* red_sx = smem;                    // [WAVES][TILE]
    float* red_sc = smem + WAVES * TILE;
    float* red_sd = smem + 2 * WAVES * TILE;
    float* terms  = smem + 3 * WAVES * TILE; // [TILE]

    // Diagonal of 16x16 f32 D: lanes 0-7 hold D[L][L] in VGPR L,
    // lanes 24-31 hold D[L-16][L-16] in VGPR L-24.
    int r = -1, v = 0;
    if (lane < 8)        { r = lane;      v = lane;      }
    else if (lane >= 24) { r = lane - 16; v = lane - 24; }
    if (r >= 0) {
        red_sx[wave * TILE + r] = acc_xx[v];
        red_sc[wave * TILE + r] = acc_cc[v];
        red_sd[wave * TILE + r] = acc_xc[v];
    }
    __syncthreads();

    if (tid < TILE) {
        float sx = 0.f, sc = 0.f, sd = 0.f;
        #pragma unroll
        for (int w = 0; w < WAVES; ++w) {
            sx += red_sx[w * TILE + tid];
            sc += red_sc[w * TILE + tid];
            sd += red_sd[w * TILE + tid];
        }
        // F.normalize semantics: v / max(||v||, eps)
        const float mx = fmaxf(sqrtf(sx), 1e-12f);
        const float mc = fmaxf(sqrtf(sc), 1e-12f);
        const float x2 = sx / (mx * mx);
        const float c2 = sc / (mc * mc);
        const float dt = sd / (mx * mc);
        const float dist = x2 + c2 - 2.0f * dt;
        terms[tid] = fminf(fmaxf(dist, 1e-12f), 1e12f);
    }
    __syncthreads();
    if (tid == 0) {
        float s = 0.f;
        #pragma unroll
        for (int i = 0; i < TILE; ++i) s += terms[i];
        partials[blockIdx.x] = s;
    }
}

// Kernel 2: deterministic tree reduce + analytic masked-out EPS term.
__global__ __launch_bounds__(THREADS)
void center_loss_reduce_kernel(const float* __restrict__ partials,
                               float* __restrict__ out, int n) {
    __shared__ float s[THREADS];
    const int t = threadIdx.x;
    float v = 0.f;
    for (int i = t; i < n; i += THREADS) v += partials[i];
    s[t] = v;
    __syncthreads();
    #pragma unroll
    for (int off = THREADS / 2; off > 0; off >>= 1) {
        if (t < off) s[t] += s[t + off];
        __syncthreads();
    }
    if (t == 0) {
        // masked-out entries: clip(0, 1e-12, 1e12) each -> B*(C-1)*1e-12 total
        double total = (double)s[0] + (double)B_SZ * (double)(C_SZ - 1) * 1e-12;
        out[0] = (float)(0.0005 * (total / (double)B_SZ));
    }
}

extern "C" void kernel_launch(void* const* d_in, const int* in_sizes, int n_in,
                              void* d_out, int out_size, void* d_ws, size_t ws_size,
                              hipStream_t stream) {
    (void)in_sizes; (void)n_in; (void)out_size; (void)ws_size;
    const float* xp      = (const float*)d_in[0];
    const int*   labels  = (const int*)d_in[1];
    const float* centers = (const float*)d_in[2];
    float*       out     = (float*)d_out;
    float*       parts   = (float*)d_ws;      // 256 floats of scratch

    const int nblocks = B_SZ / TILE;          // 256
    hipLaunchKernelGGL(center_loss_tile_kernel, dim3(nblocks), dim3(THREADS), 0,
                       stream, xp, labels, centers, parts);
    hipLaunchKernelGGL(center_loss_reduce_kernel, dim3(1), dim3(THREADS), 0,
                       stream, parts, out, nblocks);
}